// CosmopsychiaPINN_8615704396409
// MI455X (gfx1250) — compile-verified
//
#include <hip/hip_runtime.h>
#include <math.h>

typedef __attribute__((ext_vector_type(2))) float v2f;
typedef __attribute__((ext_vector_type(8))) float v8f;

#define HDIM   512
#define PWG    8            // points per workgroup
#define NWAVE  8            // waves per workgroup (wave32)
#define SROW   516          // padded state row stride (floats) -> bank-conflict-free A frags
#define SPT    (16 * SROW)  // floats per point state block (16 states x 516)
#define SCROFF (PWG * SPT)  // scratch offset in floats
#define SCRW   256          // scratch floats per wave (16x16 tile)
#define NLAYER 6

__global__ __launch_bounds__(256, 2)
void pinn_wmma_kernel(const float* __restrict__ coords,
                      const float* __restrict__ W0,
                      const float* __restrict__ b0,
                      const float* __restrict__ Wh,
                      const float* __restrict__ bh,
                      const float* __restrict__ Wo,
                      const float* __restrict__ bo,
                      const float* __restrict__ hbar_p,
                      const float* __restrict__ visc_p,
                      const float* __restrict__ coup_p,
                      float* __restrict__ out)
{
    extern __shared__ float smem[];
    float* S   = smem;            // 8 points x 16 states x 516
    float* scr = smem + SCROFF;   // 8 waves x 256

    const int tid  = threadIdx.x;
    const int lane = tid & 31;
    const int wv   = tid >> 5;
    const int l16  = lane & 15;
    const int lh   = lane >> 4;
    const int Bp   = gridDim.x * PWG;   // total points (8192)

    // ---------------- Layer 0: seed value / grad / hessian state ----------------
    {
        const int gp = blockIdx.x * PWG + wv;     // wave w seeds point w
        const float x0 = coords[gp * 4 + 0];
        const float x1 = coords[gp * 4 + 1];
        const float x2 = coords[gp * 4 + 2];
        const float x3 = coords[gp * 4 + 3];
        float* Sp = S + wv * SPT;
        for (int n = lane; n < HDIM; n += 32) {
            float w0 = W0[n], w1 = W0[HDIM + n], w2 = W0[2 * HDIM + n], w3 = W0[3 * HDIM + n];
            float z  = b0[n] + x0 * w0 + x1 * w1 + x2 * w2 + x3 * w3;
            float a  = tanhf(z);
            float s2 = 1.0f - a * a;
            float m  = -2.0f * a * s2;            // h_jk = s2*(0 - 2 a z_j z_k)
            Sp[0 * SROW + n]  = a;
            Sp[1 * SROW + n]  = s2 * w0;
            Sp[2 * SROW + n]  = s2 * w1;
            Sp[3 * SROW + n]  = s2 * w2;
            Sp[4 * SROW + n]  = s2 * w3;
            Sp[5 * SROW + n]  = m * w0 * w0;      // (0,0)
            Sp[6 * SROW + n]  = m * w0 * w1;      // (0,1)
            Sp[7 * SROW + n]  = m * w0 * w2;      // (0,2)
            Sp[8 * SROW + n]  = m * w0 * w3;      // (0,3)
            Sp[9 * SROW + n]  = m * w1 * w1;      // (1,1)
            Sp[10 * SROW + n] = m * w1 * w2;      // (1,2)
            Sp[11 * SROW + n] = m * w1 * w3;      // (1,3)
            Sp[12 * SROW + n] = m * w2 * w2;      // (2,2)
            Sp[13 * SROW + n] = m * w2 * w3;      // (2,3)
            Sp[14 * SROW + n] = m * w3 * w3;      // (3,3)
            Sp[15 * SROW + n] = 0.0f;             // pad state
        }
    }
    __syncthreads();

    // ---------------- 6 hidden layers: WMMA GEMM + in-place tanh propagation ----
    const v8f vzero = {0.f, 0.f, 0.f, 0.f, 0.f, 0.f, 0.f, 0.f};
    #pragma unroll 1
    for (int layer = 0; layer < NLAYER; ++layer) {
        const float* W = Wh + (size_t)layer * HDIM * HDIM;

        v8f acc[4][PWG];
        #pragma unroll
        for (int i = 0; i < 4; ++i)
            #pragma unroll
            for (int p = 0; p < PWG; ++p)
                acc[i][p] = vzero;

        // GEMM: Z(16x512 per point) = S(16x512) * W(512x512), K-steps of 4.
        // Outer-product order with register double-buffered B fragments:
        // B pairs for kk+1 are in flight while the 32 WMMAs of kk execute.
        {
            // B base: row (2*lh), col wv*16 + l16; tile i adds 128 floats, kk adds 2048
            const float* Wb = W + (2 * lh) * HDIM + wv * 16 + l16;
            const float* Ab = S + l16 * SROW + 2 * lh;
            v2f bf[4];
            #pragma unroll
            for (int i = 0; i < 4; ++i) {
                bf[i].x = Wb[i * (NWAVE * 16)];
                bf[i].y = Wb[i * (NWAVE * 16) + HDIM];
            }
            #pragma unroll 2
            for (int kk = 0; kk < 128; ++kk) {
                v2f af[PWG];
                #pragma unroll
                for (int p = 0; p < PWG; ++p)
                    af[p] = *(const v2f*)(Ab + p * SPT + 4 * kk);
                const int kn = (kk < 127) ? (kk + 1) : 127;   // prefetch (re-reads last)
                v2f bn[4];
                #pragma unroll
                for (int i = 0; i < 4; ++i) {
                    bn[i].x = Wb[kn * 4 * HDIM + i * (NWAVE * 16)];
                    bn[i].y = Wb[kn * 4 * HDIM + i * (NWAVE * 16) + HDIM];
                }
                #pragma unroll
                for (int i = 0; i < 4; ++i)
                    #pragma unroll
                    for (int p = 0; p < PWG; ++p)
                        acc[i][p] = __builtin_amdgcn_wmma_f32_16x16x4_f32(
                            false, af[p], false, bf[i], (short)0, acc[i][p], false, false);
                #pragma unroll
                for (int i = 0; i < 4; ++i)
                    bf[i] = bn[i];
            }
        }
        __syncthreads();    // all K-reads of S complete before overwrite

        // Activation: stage each 16x16 Z tile through wave scratch, rewrite S
        float* zs = scr + wv * SCRW;
        #pragma unroll
        for (int i = 0; i < 4; ++i) {
            const int n0 = (wv + NWAVE * i) * 16;
            const float bias = bh[layer * HDIM + n0 + l16];
            #pragma unroll
            for (int p = 0; p < PWG; ++p) {
                v8f t = acc[i][p];
                #pragma unroll
                for (int v = 0; v < 8; ++v)               // D layout: M = v + 8*lh, N = l16
                    zs[(v + 8 * lh) * 16 + l16] = t[v];
                asm volatile("s_wait_dscnt 0" ::: "memory");
                const int c = l16;
                float z0 = zs[c] + bias;                  // bias only on value row
                float a  = tanhf(z0);
                float s2 = 1.0f - a * a;
                float g0 = zs[16 + c], g1 = zs[32 + c], g2 = zs[48 + c], g3 = zs[64 + c];
                float* Sp = S + p * SPT + n0 + c;
                if (lh == 0) {                            // states 0..7
                    Sp[0]        = a;
                    Sp[SROW * 1] = s2 * g0;
                    Sp[SROW * 2] = s2 * g1;
                    Sp[SROW * 3] = s2 * g2;
                    Sp[SROW * 4] = s2 * g3;
                    Sp[SROW * 5] = s2 * (zs[5 * 16 + c] - 2.0f * a * g0 * g0);
                    Sp[SROW * 6] = s2 * (zs[6 * 16 + c] - 2.0f * a * g0 * g1);
                    Sp[SROW * 7] = s2 * (zs[7 * 16 + c] - 2.0f * a * g0 * g2);
                } else {                                  // states 8..15
                    Sp[SROW * 8]  = s2 * (zs[8 * 16 + c]  - 2.0f * a * g0 * g3);
                    Sp[SROW * 9]  = s2 * (zs[9 * 16 + c]  - 2.0f * a * g1 * g1);
                    Sp[SROW * 10] = s2 * (zs[10 * 16 + c] - 2.0f * a * g1 * g2);
                    Sp[SROW * 11] = s2 * (zs[11 * 16 + c] - 2.0f * a * g1 * g3);
                    Sp[SROW * 12] = s2 * (zs[12 * 16 + c] - 2.0f * a * g2 * g2);
                    Sp[SROW * 13] = s2 * (zs[13 * 16 + c] - 2.0f * a * g2 * g3);
                    Sp[SROW * 14] = s2 * (zs[14 * 16 + c] - 2.0f * a * g3 * g3);
                    Sp[SROW * 15] = 0.0f;
                }
                asm volatile("s_wait_dscnt 0" ::: "memory");  // tile writes before zs reuse
            }
        }
        __syncthreads();
    }

    // ---------------- Output head (512x5) + physics epilogue ----------------
    {
        float* zs = scr + wv * SCRW;
        const float* Sp = S + wv * SPT;                  // wave w finishes point w
        for (int base = 0; base < 80; base += 32) {
            int id = base + lane;
            if (id < 80) {
                int s = id / 5, o = id % 5;
                float av = (s == 0) ? bo[o] : 0.0f;
                for (int n = 0; n < HDIM; ++n)
                    av += Sp[s * SROW + n] * Wo[n * 5 + o];
                zs[id] = av;                              // O[s][o]
            }
        }
        asm volatile("s_wait_dscnt 0" ::: "memory");
        if (lane == 0) {
            const float hb  = hbar_p[0];
            const float vis = visc_p[0];
            const float cp  = coup_p[0];
            const int   P   = blockIdx.x * PWG + wv;
            #define O(s, o) zs[(s) * 5 + (o)]
            float pr = O(0, 0), pi = O(0, 1);
            float u0 = O(0, 2), u1 = O(0, 3), u2 = O(0, 4);
            float lap0 = O(5, 0) + O(9, 0) + O(12, 0);
            float lap1 = O(5, 1) + O(9, 1) + O(12, 1);
            float lap2 = O(5, 2) + O(9, 2) + O(12, 2);
            float lap3 = O(5, 3) + O(9, 3) + O(12, 3);
            float lap4 = O(5, 4) + O(9, 4) + O(12, 4);
            float dt0 = O(4, 0), dt1 = O(4, 1), dt2 = O(4, 2), dt3 = O(4, 3), dt4 = O(4, 4);
            float hh = 0.5f * hb * hb;
            float qr = -hb * dt1 + hh * lap0;
            float qi =  hb * dt0 + hh * lap1;
            float rho = pr * pr + pi * pi;
            float conv0 = u0 * O(1, 2) + u1 * O(2, 2) + u2 * O(3, 2);
            float conv1 = u0 * O(1, 3) + u1 * O(2, 3) + u2 * O(3, 3);
            float conv2 = u0 * O(1, 4) + u1 * O(2, 4) + u2 * O(3, 4);
            float pc0 = pr * O(1, 0) + pi * O(1, 1);
            float pc1 = pr * O(2, 0) + pi * O(2, 1);
            float pc2 = pr * O(3, 0) + pi * O(3, 1);
            float ns0 = dt2 + conv0 - vis * lap2 - cp * (pc0 - u0 * rho);
            float ns1 = dt3 + conv1 - vis * lap3 - cp * (pc1 - u1 * rho);
            float ns2 = dt4 + conv2 - vis * lap4 - cp * (pc2 - u2 * rho);
            float dv  = O(1, 2) + O(2, 3) + O(3, 4);
            float ph  = atan2f(pi, pr);
            #undef O
            out[2 * P]     = pr;   out[2 * P + 1] = pi;      // psi (complex64)
            float* ou = out + 2 * Bp;
            ou[3 * P] = u0; ou[3 * P + 1] = u1; ou[3 * P + 2] = u2;
            float* oq = out + 5 * Bp;
            oq[2 * P] = qr; oq[2 * P + 1] = qi;              // quantum (complex64)
            float* on = out + 7 * Bp;
            on[3 * P] = ns0; on[3 * P + 1] = ns1; on[3 * P + 2] = ns2;
            out[10 * Bp + P] = dv;
            out[11 * Bp + P] = rho;
            out[12 * Bp + P] = ph;
            float* op = out + 13 * Bp;
            op[3 * P] = pc0; op[3 * P + 1] = pc1; op[3 * P + 2] = pc2;
        }
    }
}

extern "C" void kernel_launch(void* const* d_in, const int* in_sizes, int n_in,
                              void* d_out, int out_size, void* d_ws, size_t ws_size,
                              hipStream_t stream) {
    (void)n_in; (void)out_size; (void)d_ws; (void)ws_size;
    const int npts  = in_sizes[0] / 4;              // coordinates: (B,4)
    const int grid  = npts / PWG;
    const size_t shmem = (size_t)(SCROFF + NWAVE * SCRW) * sizeof(float);  // ~272 KB
    (void)hipFuncSetAttribute((const void*)pinn_wmma_kernel,
                              hipFuncAttributeMaxDynamicSharedMemorySize, (int)shmem);
    pinn_wmma_kernel<<<grid, 256, shmem, stream>>>(
        (const float*)d_in[0],  // coordinates
        (const float*)d_in[1],  // W0
        (const float*)d_in[2],  // b0
        (const float*)d_in[3],  // Wh
        (const float*)d_in[4],  // bh
        (const float*)d_in[5],  // Wo
        (const float*)d_in[6],  // bo
        (const float*)d_in[7],  // hbar
        (const float*)d_in[8],  // viscosity
        (const float*)d_in[9],  // coupling
        (float*)d_out);
}